// _HSTU_BSA_Triton_23622320128062
// MI455X (gfx1250) — compile-verified
//
#include <hip/hip_runtime.h>
#include <math.h>
#include <limits.h>

// HSTU block-sparse attention forward for MI455X (gfx1250, wave32).
// fp32 WMMA (v_wmma_f32_16x16x4_f32) for the compressed-attention GEMMs to
// keep exact fp32 scores (top-k selection is discrete -> precision-critical).
// Masked operands are produced branch-free (clamped row + 0/1 multiply) so
// fragment loads batch ahead of the WMMA chain instead of serializing.

typedef __attribute__((ext_vector_type(2))) float v2f;
typedef __attribute__((ext_vector_type(4))) float v4f;
typedef __attribute__((ext_vector_type(8))) float v8f;

#define HH   4      // heads
#define DD   64     // head dim
#define HD   256    // HH*DD
#define BS   32     // compression block size
#define STOP 4      // top-k blocks selected

__device__ __forceinline__ v8f wmma4(v2f a, v2f b, v8f c) {
  // D = A(16x4 f32) * B(4x16 f32) + C(16x16 f32)
  return __builtin_amdgcn_wmma_f32_16x16x4_f32(false, a, false, b, (short)0, c,
                                               false, false);
}

__device__ __forceinline__ float silu_f(float x) {
  return x / (1.0f + expf(-x));
}

// ---------------------------------------------------------------------------
// Kernel 1: block mean-pool k,v -> k_cmp,v_cmp.  grid=(Tcmax,H), block=64 (=D)
// ---------------------------------------------------------------------------
__global__ __launch_bounds__(64) void k_compress(
    const float* __restrict__ k, const float* __restrict__ v,
    const long long* __restrict__ offs,
    float* __restrict__ kc, float* __restrict__ vc, int B)
{
  const int c = blockIdx.x, h = blockIdx.y, d = threadIdx.x;
  int cum = 0, start = -1, cnt = 0;
  for (int i = 0; i < B; ++i) {
    const int s = (int)offs[i], e = (int)offs[i + 1];
    const int cl = (e - s + BS - 1) >> 5;
    if (c >= cum && c < cum + cl) {
      const int loc = c - cum;
      start = s + loc * BS;
      cnt = min(BS, e - s - loc * BS);
    }
    cum += cl;
  }
  float ks = 0.f, vs = 0.f;
  if (start >= 0) {
    for (int i = 0; i < cnt; ++i) {
      const size_t base = (size_t)(start + i) * HD + h * DD + d;
      ks += k[base];
      vs += v[base];
    }
    const float inv = 1.0f / (float)(cnt > 0 ? cnt : 1);
    ks *= inv; vs *= inv;
  }
  const size_t o = (size_t)c * HD + h * DD + d;
  kc[o] = ks;   // out-of-range compressed rows are zeroed (safe to gather)
  vc[o] = vs;
}

// ---------------------------------------------------------------------------
// Kernel 2: compressed attention + top-4 selection.
// One wave32 per (16-token tile, head).  WMMA f32 16x16x4.
// grid=((T+15)/16, H), block=32
// ---------------------------------------------------------------------------
__global__ __launch_bounds__(32) void k_cmp_attn(
    const float* __restrict__ q, const float* __restrict__ g_cmp,
    const float* __restrict__ kc, const float* __restrict__ vc,
    const long long* __restrict__ offs,
    float* __restrict__ o_cmp, int* __restrict__ topk,
    int T, int B)
{
  __shared__ float pLds[16 * 17];  // silu(P) tile, stride 17 (bank-conflict pad)
  __shared__ float sLds[16 * 16];  // raw selection scores

  const int h  = blockIdx.y;
  const int t0 = blockIdx.x * 16;
  if (t0 >= T) return;
  const int lane = threadIdx.x;
  const int lo = lane & 15, hi = lane >> 4;

  // locate batch of this tile (tiles never straddle batches: offsets % 32 == 0)
  int off_b = 0, off_e = T, cmp_off = 0, cmp_len = 1;
  {
    int cc = 0;
    for (int i = 0; i < B; ++i) {
      const int s = (int)offs[i], e = (int)offs[i + 1];
      const int cl = (e - s + BS - 1) >> 5;
      if (t0 >= s && t0 < e) { off_b = s; off_e = e; cmp_off = cc; cmp_len = cl; }
      cc += cl;
    }
  }
  const int p0 = t0 - off_b;  // local position of first row

  // Q fragments (A-matrix layout): lane holds row lo, K-pair 4kk+2hi
  const int tq = min(t0 + lo, off_e - 1);
  const float* qrow = q + (size_t)tq * HD + h * DD;
  v2f aq[16];
#pragma unroll
  for (int kk = 0; kk < 16; ++kk)
    aq[kk] = *(const v2f*)(qrow + 4 * kk + 2 * hi);

  v8f acco[4];
#pragma unroll
  for (int dt = 0; dt < 4; ++dt)
#pragma unroll
    for (int r = 0; r < 8; ++r) acco[dt][r] = 0.f;

  float tv[STOP]; int ti[STOP];
#pragma unroll
  for (int s = 0; s < STOP; ++s) { tv[s] = -__builtin_huge_valf(); ti[s] = INT_MAX; }

  const int nchunk = (cmp_len + 15) >> 4;
  for (int ch = 0; ch < nchunk; ++ch) {
    // ---- scores = Q * Kc^T over K=64 (16 chained wmma steps) ----
    // Branch-free masked B fragments: clamp row (always valid, finite) and
    // multiply by 0/1 so all 16 loads issue back-to-back before the WMMAs.
    const int  j   = ch * 16 + lo;       // B-matrix column = block index
    const float mj = (j < cmp_len) ? 1.0f : 0.0f;
    const int  crow = cmp_off + (j < cmp_len ? j : 0);
    const float* kcr = kc + (size_t)crow * HD + h * DD;

    v2f bfr[16];
#pragma unroll
    for (int kk = 0; kk < 16; ++kk)
      bfr[kk] = *(const v2f*)(kcr + 4 * kk + 2 * hi);

    v8f acc;
#pragma unroll
    for (int r = 0; r < 8; ++r) acc[r] = 0.f;
#pragma unroll
    for (int kk = 0; kk < 16; ++kk)
      acc = wmma4(aq[kk], bfr[kk] * mj, acc);

    // ---- scale, causal mask, silu; stash P and selection scores in LDS ----
#pragma unroll
    for (int r = 0; r < 8; ++r) {
      const int m   = r + 8 * hi;        // D-layout row
      const int tok = t0 + m;
      const bool rv = tok < off_e;
      const float sc = acc[r] * 0.125f;  // scale = D^-0.5 = 1/8
      const int qblk = (p0 + m) >> 5;
      const bool causal = rv && (j <= qblk) && (j < cmp_len);
      pLds[m * 17 + lo] = causal ? silu_f(sc) : 0.f;
      sLds[m * 16 + lo] = causal ? sc : -__builtin_huge_valf();
    }
    __syncthreads();

    // ---- stable top-4 merge (ties -> lowest index; matches jax top_k) ----
    if (lane < 16) {
      for (int n = 0; n < 16; ++n) {
        const float vv = sLds[lane * 16 + n];
        const int jj = ch * 16 + n;
        int pos = STOP;
        while (pos > 0 &&
               (vv > tv[pos - 1] || (vv == tv[pos - 1] && jj < ti[pos - 1])))
          pos--;
        if (pos < STOP) {
          for (int x = STOP - 1; x > pos; --x) { tv[x] = tv[x - 1]; ti[x] = ti[x - 1]; }
          tv[pos] = vv; ti[pos] = jj;
        }
      }
    }

    // ---- o_cmp += P(16x16) * Vc(16x64): 4 column tiles x 4 K-steps ----
    // Preload all Vc fragments branch-free (clamped rows + 0/1 masks).
    float bvx[4][4], bvy[4][4];
#pragma unroll
    for (int kk = 0; kk < 4; ++kk) {
      const int j0 = ch * 16 + 4 * kk + 2 * hi;
      const int j1 = j0 + 1;
      const float m0 = (j0 < cmp_len) ? 1.0f : 0.0f;
      const float m1 = (j1 < cmp_len) ? 1.0f : 0.0f;
      const int r0 = cmp_off + (j0 < cmp_len ? j0 : 0);
      const int r1 = cmp_off + (j1 < cmp_len ? j1 : 0);
      const float* p0v = vc + (size_t)r0 * HD + h * DD + lo;
      const float* p1v = vc + (size_t)r1 * HD + h * DD + lo;
#pragma unroll
      for (int dt = 0; dt < 4; ++dt) {
        bvx[kk][dt] = p0v[dt * 16] * m0;
        bvy[kk][dt] = p1v[dt * 16] * m1;
      }
    }
    v2f ap[4];
#pragma unroll
    for (int kk = 0; kk < 4; ++kk) {
      const int kb = 4 * kk + 2 * hi;
      ap[kk].x = pLds[lo * 17 + kb];
      ap[kk].y = pLds[lo * 17 + kb + 1];
    }
#pragma unroll
    for (int dt = 0; dt < 4; ++dt) {
#pragma unroll
      for (int kk = 0; kk < 4; ++kk) {
        v2f bv; bv.x = bvx[kk][dt]; bv.y = bvy[kk][dt];
        acco[dt] = wmma4(ap[kk], bv, acco[dt]);
      }
    }
    __syncthreads();  // LDS reused next chunk
  }

  // ---- write top-k indices ----
  if (lane < 16) {
    const int tok = t0 + lane;
    if (tok < off_e) {
#pragma unroll
      for (int s = 0; s < STOP; ++s)
        topk[((size_t)tok * HH + h) * STOP + s] = ti[s];
    }
  }
  // ---- write o_cmp (D-layout: row = r+8*hi, dim = dt*16+lo) ----
#pragma unroll
  for (int dt = 0; dt < 4; ++dt) {
#pragma unroll
    for (int r = 0; r < 8; ++r) {
      const int m = r + 8 * hi;
      const int tok = t0 + m;
      if (tok < off_e) {
        o_cmp[(size_t)tok * HD + h * DD + dt * 16 + lo] =
            acco[dt][r] * g_cmp[tok * HH + h];
      }
    }
  }
}

// ---------------------------------------------------------------------------
// Kernel 3: selected (gathered) attention.  One wave32 per (token, head).
// Per-token gather -> no shared B matrix, so plain VALU GEMV.
// grid=(T,H), block=32
// ---------------------------------------------------------------------------
__global__ __launch_bounds__(32) void k_slc_attn(
    const float* __restrict__ q, const float* __restrict__ k,
    const float* __restrict__ v, const float* __restrict__ g_slc,
    const long long* __restrict__ offs, const int* __restrict__ topk,
    float* __restrict__ o_slc, int T, int B)
{
  __shared__ __align__(16) float qL[DD];
  __shared__ float pL[STOP * BS];
  __shared__ int   tgtL[STOP * BS];

  const int t = blockIdx.x, h = blockIdx.y, lane = threadIdx.x;
  long long off_b = 0;
  for (int i = 0; i < B; ++i)
    if ((long long)t >= offs[i] && (long long)t < offs[i + 1]) off_b = offs[i];

  const float* qrow = q + (size_t)t * HD + h * DD;
  qL[2 * lane]     = qrow[2 * lane];
  qL[2 * lane + 1] = qrow[2 * lane + 1];

  int bi[STOP];
  const int* tkp = topk + ((size_t)t * HH + h) * STOP;
#pragma unroll
  for (int s = 0; s < STOP; ++s) bi[s] = tkp[s];
  __syncthreads();

  // scores: 4 gathered keys per lane (128 keys total)
#pragma unroll
  for (int ii = 0; ii < 4; ++ii) {
    const int key = lane * 4 + ii;
    const int s = key >> 5, i = key & 31;
    const long long tg = off_b + (long long)bi[s] * BS + i;
    const bool causal = tg <= (long long)t;  // token-level causal mask
    long long tgl = tg;
    if (tgl < 0) tgl = 0;
    if (tgl > (long long)(T - 1)) tgl = T - 1;
    const int tgc = (int)tgl;
    const float* krow = k + (size_t)tgc * HD + h * DD;
    float sc = 0.f;
#pragma unroll
    for (int d = 0; d < DD; d += 4) {
      const v4f qq = *(const v4f*)(qL + d);
      const v4f kk4 = *(const v4f*)(krow + d);
      sc += qq.x * kk4.x + qq.y * kk4.y + qq.z * kk4.z + qq.w * kk4.w;
    }
    sc *= 0.125f;
    pL[key] = causal ? silu_f(sc) : 0.f;
    tgtL[key] = tgc;
  }
  __syncthreads();

  // output: each lane owns 2 dims (v2f loads), reduce over 128 gathered keys
  v2f o2; o2.x = 0.f; o2.y = 0.f;
  const int d0 = 2 * lane;
  for (int key = 0; key < STOP * BS; ++key) {
    const float p = pL[key];
    const v2f vv = *(const v2f*)(v + (size_t)tgtL[key] * HD + h * DD + d0);
    o2 += vv * p;
  }
  const float g = g_slc[t * HH + h];
  float* orow = o_slc + (size_t)t * HD + h * DD + d0;
  orow[0] = o2.x * g;
  orow[1] = o2.y * g;
}

// ---------------------------------------------------------------------------
extern "C" void kernel_launch(void* const* d_in, const int* in_sizes, int n_in,
                              void* d_out, int out_size, void* d_ws, size_t ws_size,
                              hipStream_t stream)
{
  const float*     q      = (const float*)d_in[0];
  const float*     k      = (const float*)d_in[1];
  const float*     v      = (const float*)d_in[2];
  const float*     g_cmp  = (const float*)d_in[3];
  const float*     g_slc  = (const float*)d_in[4];
  const long long* offs   = (const long long*)d_in[5];

  const int B = in_sizes[5] - 1;
  const int T = in_sizes[0] / HD;          // (T,H,D) with H=4, D=64
  const int Tcmax = T / BS + B + 1;        // upper bound on compressed rows

  float* kc   = (float*)d_ws;
  float* vc   = kc + (size_t)Tcmax * HD;
  int*   topk = (int*)(vc + (size_t)Tcmax * HD);

  float* o_cmp = (float*)d_out;
  float* o_slc = o_cmp + (size_t)T * HD;

  k_compress<<<dim3(Tcmax, HH), 64, 0, stream>>>(k, v, offs, kc, vc, B);
  k_cmp_attn<<<dim3((T + 15) / 16, HH), 32, 0, stream>>>(
      q, g_cmp, kc, vc, offs, o_cmp, topk, T, B);
  k_slc_attn<<<dim3(T, HH), 32, 0, stream>>>(
      q, k, v, g_slc, offs, topk, o_slc, T, B);
}